// PFSA_41343355191549
// MI455X (gfx1250) — compile-verified
//
#include <hip/hip_runtime.h>
#include <math.h>

typedef __attribute__((ext_vector_type(2))) float v2f;
typedef __attribute__((ext_vector_type(4))) float v4f;
typedef __attribute__((ext_vector_type(8))) float v8f;

constexpr int T   = 4096;
constexpr int F   = 1024;
constexpr int NCH = 32;        // T-chunks for pass 1
constexpr int TCH = T / NCH;   // 128 timesteps per chunk

// ---------------- Pass 1: masked partial column sums over T ----------------
__global__ __launch_bounds__(256) void pfsa_partial(const float* __restrict__ x,
                                                    const int*   __restrict__ mask,
                                                    float*       __restrict__ part) {
  const int b = blockIdx.y, ch = blockIdx.x, tid = threadIdx.x;
  const float* xb = x + ((size_t)(b * T + ch * TCH)) * F;
  const int*   mb = mask + (size_t)b * T + ch * TCH;
  v4f acc = {0.f, 0.f, 0.f, 0.f};
  for (int t = 0; t < TCH; ++t) {
    if (mb[t] != 0) {                                   // uniform branch: skip padded rows
      acc += *((const v4f*)(xb + (size_t)t * F) + tid); // coalesced b128
    }
  }
  *((v4f*)(part + ((size_t)(b * NCH + ch)) * F) + tid) = acc;
}

// ---------------- Pass 2: finalize k -> k_centered, kd, sum(k_c), per batch ----------------
__device__ __forceinline__ float block_reduce256(float v, float* red, int tid) {
  __syncthreads();                 // protect previous use of red[]
  red[tid] = v;
  __syncthreads();
  for (int s = 128; s > 0; s >>= 1) {
    if (tid < s) red[tid] += red[tid + s];
    __syncthreads();
  }
  return red[0];
}

__global__ __launch_bounds__(256) void pfsa_finalize(const float* __restrict__ part,
                                                     const int*   __restrict__ mask,
                                                     float*       __restrict__ kc,
                                                     float*       __restrict__ stats) {
  __shared__ float red[256];
  const int b = blockIdx.x, tid = threadIdx.x;

  // valid length L[b]
  int ls = 0;
  for (int t = tid; t < T; t += 256) ls += mask[(size_t)b * T + t];
  const float L = block_reduce256((float)ls, red, tid);

  // k = (sum over chunks) / L   (4 channels per thread)
  v4f k4 = {0.f, 0.f, 0.f, 0.f};
  for (int c = 0; c < NCH; ++c)
    k4 += *((const v4f*)(part + ((size_t)(b * NCH + c)) * F) + tid);
  k4 *= (1.0f / L);

  // center over channels
  const float ksum = block_reduce256(k4.x + k4.y + k4.z + k4.w, red, tid);
  const float mean = ksum * (1.0f / (float)F);
  v4f kc4 = k4 - mean;
  *((v4f*)(kc + (size_t)b * F) + tid) = kc4;

  const float kd2   = block_reduce256(kc4.x*kc4.x + kc4.y*kc4.y + kc4.z*kc4.z + kc4.w*kc4.w, red, tid);
  const float sumkc = block_reduce256(kc4.x + kc4.y + kc4.z + kc4.w, red, tid);
  if (tid == 0) {
    stats[b * 2 + 0] = sqrtf(kd2);
    stats[b * 2 + 1] = sumkc;
  }
}

// ---------------- Pass 3: per-row stats via WMMA f32 16x16x4, scale + store ----------------
// A (16x4) = 16 timesteps x 4-wide K slice of x. B (4x16): col0 = k_c slice, col1 = ones.
// Accumulator col0 -> dot(x,k_c), col1 -> rowsum(x), for all 16 rows of the tile at once.
__global__ __launch_bounds__(256) void pfsa_main(const float* __restrict__ x,
                                                 const int*   __restrict__ mask,
                                                 const float* __restrict__ kc,
                                                 const float* __restrict__ stats,
                                                 float*       __restrict__ out) {
  __shared__ float sDot[128], sS1[128], sS2[128];
  const int tid  = threadIdx.x;
  const int lane = tid & 31;
  const int wave = tid >> 5;
  const int b    = blockIdx.y;
  const int t0   = blockIdx.x * 128 + wave * 16;   // 16-row tile per wave
  const int r    = lane & 15;                      // tile row (A) / column N (B)
  const int kh   = (lane >> 4) << 1;               // K sub-offset: lanes 0-15 -> K0,1; 16-31 -> K2,3

  const float* xrow = x  + ((size_t)(b * T + t0 + r)) * F + kh;
  const float* kcb  = kc + (size_t)b * F + kh;

  v8f   acc = {};
  float s2  = 0.f;
  for (int k0 = 0; k0 < F; k0 += 4) {
    v2f a  = *(const v2f*)(xrow + k0);   // A tile: row r, K = k0+kh, k0+kh+1
    v2f kv = *(const v2f*)(kcb  + k0);   // broadcast k_c slice
    s2 = __builtin_fmaf(a.x, a.x, s2);
    s2 = __builtin_fmaf(a.y, a.y, s2);
    v2f bv;                              // B: N==0 -> k_c, N==1 -> ones, else 0
    bv.x = (r == 0) ? kv.x : ((r == 1) ? 1.0f : 0.0f);
    bv.y = (r == 0) ? kv.y : ((r == 1) ? 1.0f : 0.0f);
    acc = __builtin_amdgcn_wmma_f32_16x16x4_f32(false, a, false, bv, (short)0, acc,
                                                false, false);
  }

  // sum of squares: lane L holds half the K range of row L%16; combine with lane L^16
  s2 += __shfl_xor(s2, 16, 32);
  if (lane < 16) sS2[wave * 16 + lane] = s2;

  // D layout: VGPR j, lanes 0-15 -> M=j, N=lane; lanes 16-31 -> M=8+j, N=lane-16
  if (r < 2) {
    const int rowbase = wave * 16 + ((lane >> 4) << 3);
    float* dst = (r == 0) ? sDot : sS1;
#pragma unroll
    for (int j = 0; j < 8; ++j) dst[rowbase + j] = acc[j];
  }
  __syncthreads();

  const float kd    = stats[b * 2 + 0];
  const float sumkc = stats[b * 2 + 1];
  for (int rr = 0; rr < 16; ++rr) {
    const int   row  = wave * 16 + rr;
    const int   t    = t0 + rr;
    const float S1   = sS1[row];
    const float S2v  = sS2[row];
    const float dot  = sDot[row];
    const float mean = S1 * (1.0f / (float)F);
    const float num  = dot - mean * sumkc;                 // x_c . k_c
    const float qd   = sqrtf(fmaxf(S2v - S1 * mean, 0.f)); // ||x_c||
    const float Cqk  = num / (qd * kd);
    const float A    = 1.0f / (1.0f + __expf(Cqk));        // 1 - sigmoid(C)
    const float amp  = A * (float)mask[(size_t)b * T + t];
    const float* xr   = x   + ((size_t)(b * T + t)) * F;
    float*       orow = out + ((size_t)(b * T + t)) * F;
#pragma unroll
    for (int c = lane * 4; c < F; c += 128) {              // coalesced b128, NT stores
      v4f v = *(const v4f*)(xr + c);
      v *= amp;
      __builtin_nontemporal_store(v, (v4f*)(orow + c));
    }
  }
}

extern "C" void kernel_launch(void* const* d_in, const int* in_sizes, int n_in,
                              void* d_out, int out_size, void* d_ws, size_t ws_size,
                              hipStream_t stream) {
  (void)n_in; (void)out_size; (void)ws_size;
  const float* x    = (const float*)d_in[0];
  const int*   mask = (const int*)d_in[1];
  float*       out  = (float*)d_out;
  const int B = in_sizes[1] / T;   // attention_mask is [B,T]

  float* part  = (float*)d_ws;                        // B*NCH*F floats (~1 MB)
  float* kcbuf = part + (size_t)B * NCH * F;          // B*F floats
  float* stats = kcbuf + (size_t)B * F;               // B*2 floats

  dim3 g1(NCH, B);
  pfsa_partial<<<g1, 256, 0, stream>>>(x, mask, part);
  pfsa_finalize<<<B, 256, 0, stream>>>(part, mask, kcbuf, stats);
  dim3 g3(T / 128, B);
  pfsa_main<<<g3, 256, 0, stream>>>(x, mask, kcbuf, stats, out);
}